// MemoryModule_15247133900895
// MI455X (gfx1250) — compile-verified
//
#include <hip/hip_runtime.h>
#include <hip/hip_bf16.h>
#include <math.h>

#define HID   1024
#define MSZ   4096
#define WORDL 128
#define RH    4
#define NWW   2
#define EPSC  0.001f

typedef float v2f __attribute__((ext_vector_type(2)));
typedef float v8f __attribute__((ext_vector_type(8)));

#if defined(__HIP_DEVICE_COMPILE__) && __has_builtin(__builtin_amdgcn_wmma_f32_16x16x4_f32)
#define HAVE_WMMA_F32X4 1
#else
#define HAVE_WMMA_F32X4 0
#endif

// ---------------- scratch layout (floats) ----------------
// proj: wv[256]@0, ev[256]@256, fg[4]@512, rm[20]@516, wk[256]@536,
//       ws[2]@792, rk[512]@794, rs[4]@1306   (total 1310)
#define OFF_PROJ  0
#define OFF_NORMS 1310      // [0..1] ||wk||+EPS, [2..5] ||rk||
#define OFF_DOTW  1316      // 2*4096
#define OFF_AN    9508      // 4096  (||mem0 row||+EPS)
#define OFF_MEMN  13604     // 4096  (||memory row||, no EPS)
#define OFF_DOTR  17700     // 4*4096
#define OFF_CW    34084     // 4*4096
#define OFF_FW    50468     // 2*4*4096  [w][r][m]
#define OFF_BW    83236     // 2*4*4096  [w][r][m]
#define OFF_RW0P  116004    // 16*4096 zero-padded rw0   (rows 4..15 = 0)
#define OFF_RWTP  181540    // 16*4096 zero-padded read_weights

// ---------------- output layout (floats) ----------------
#define O_RWORDS 0            // 4*128
#define O_MEM    512          // 4096*128
#define O_RW     524800       // 4*4096
#define O_WW     541184       // 2*4096
#define O_LINK   549376       // 2*4096*4096
#define O_PREC   34103808     // 2*4096
#define O_USAGE  34112000     // 4096

__device__ __forceinline__ float waveSum(float v) {
  for (int o = 16; o > 0; o >>= 1) v += __shfl_down(v, o, 32);
  return v;
}

template<int NT>
__device__ __forceinline__ float blockSum(float v, float* red, int tid) {
  red[tid] = v; __syncthreads();
  for (int s = NT / 2; s > 0; s >>= 1) { if (tid < s) red[tid] += red[tid + s]; __syncthreads(); }
  float r = red[0]; __syncthreads();
  return r;
}
template<int NT>
__device__ __forceinline__ float blockMax(float v, float* red, int tid) {
  red[tid] = v; __syncthreads();
  for (int s = NT / 2; s > 0; s >>= 1) { if (tid < s) red[tid] = fmaxf(red[tid], red[tid + s]); __syncthreads(); }
  float r = red[0]; __syncthreads();
  return r;
}
__device__ __forceinline__ float sigm(float x) { return 1.0f / (1.0f + expf(-x)); }

// ---- 0) projections: out[row] = act(x . W[row] + b[row]) ----
__global__ void k_linear(const float* __restrict__ W, const float* __restrict__ b,
                         const float* __restrict__ x, float* __restrict__ out, int act) {
  __shared__ float red[256];
  int row = blockIdx.x, tid = threadIdx.x;
  const float* wr = W + (size_t)row * HID;
  float acc = 0.f;
  for (int j = tid; j < HID; j += 256) acc += wr[j] * x[j];
  float s = blockSum<256>(acc, red, tid);
  if (tid == 0) { float v = s + b[row]; out[row] = act ? sigm(v) : v; }
}

// ---- 1) key norms ----
__global__ void k_norms(const float* __restrict__ proj, float* __restrict__ norms) {
  __shared__ float red[128];
  int b = blockIdx.x, tid = threadIdx.x;
  const float* v = (b < 2) ? (proj + 536 + b * WORDL) : (proj + 794 + (b - 2) * WORDL);
  float x = v[tid];
  float s = blockSum<128>(x * x, red, tid);
  if (tid == 0) norms[b] = sqrtf(s) + (b < 2 ? EPSC : 0.f);
}

// ---- 2) usage ----
__global__ void k_usage(const float* __restrict__ usage0, const float* __restrict__ ww0,
                        const float* __restrict__ rw0, const float* __restrict__ proj,
                        float* __restrict__ out_usage) {
  int m = blockIdx.x * 256 + threadIdx.x;
  if (m >= MSZ) return;
  float pw = ww0[m] * ww0[MSZ + m];
  float u = usage0[m] + (1.f - usage0[m]) * (1.f - pw);
  float phi = 1.f;
  for (int r = 0; r < RH; r++) phi *= proj[512 + r] * rw0[(size_t)r * MSZ + m];
  out_usage[m] = u * phi;
}

// ---- 3) dot_w + a_n (one wave per memory row) ----
__global__ void k_dotw(const float* __restrict__ mem0, const float* __restrict__ proj,
                       float* __restrict__ dotw, float* __restrict__ an) {
  __shared__ float swk[256];
  int tid = threadIdx.x;
  swk[tid] = proj[536 + tid];
  __syncthreads();
  int wave = tid >> 5, lane = tid & 31;
  int m = blockIdx.x * 8 + wave;
  const float* row = mem0 + (size_t)m * WORDL;
  float a0 = 0.f, a1 = 0.f, an2 = 0.f;
  for (int k = lane; k < WORDL; k += 32) {
    float v = row[k];
    a0 += v * swk[k]; a1 += v * swk[128 + k]; an2 += v * v;
  }
  a0 = waveSum(a0); a1 = waveSum(a1); an2 = waveSum(an2);
  if (lane == 0) { dotw[m] = a0; dotw[MSZ + m] = a1; an[m] = sqrtf(an2) + EPSC; }
}

// ---- 4) write-content softmax -> ww ----
__global__ void k_ww(const float* __restrict__ dotw, const float* __restrict__ an,
                     const float* __restrict__ norms, const float* __restrict__ proj,
                     float* __restrict__ ww) {
  __shared__ float red[1024];
  int w = blockIdx.x, tid = threadIdx.x;
  float bn = norms[w], wsw = proj[792 + w];
  float s[4], lmax = -3.4e38f;
  for (int j = 0; j < 4; j++) {
    int m = tid + j * 1024;
    float sc = dotw[(size_t)w * MSZ + m] / (128.f * (bn * an[m]) + EPSC) * wsw;
    s[j] = sc; lmax = fmaxf(lmax, sc);
  }
  float gmax = blockMax<1024>(lmax, red, tid);
  float e[4], lsum = 0.f;
  for (int j = 0; j < 4; j++) { e[j] = expf(s[j] - gmax); lsum += e[j]; }
  float gsum = blockSum<1024>(lsum, red, tid);
  for (int j = 0; j < 4; j++) ww[(size_t)w * MSZ + tid + j * 1024] = e[j] / gsum;
}

// ---- 5) memory update + row norms ----
__global__ void k_memory(const float* __restrict__ mem0, const float* __restrict__ ww,
                         const float* __restrict__ proj, float* __restrict__ memo,
                         float* __restrict__ memn) {
  __shared__ float red[128];
  int m = blockIdx.x, k = threadIdx.x;
  float w0 = ww[m], w1 = ww[MSZ + m];
  float ev0 = proj[256 + k], ev1 = proj[256 + 128 + k];
  float wv0 = proj[k], wv1 = proj[128 + k];
  float mv = mem0[(size_t)m * WORDL + k] * ((1.f - w0 * ev0) * (1.f - w1 * ev1))
           + w0 * wv0 + w1 * wv1;
  memo[(size_t)m * WORDL + k] = mv;
  float s = blockSum<128>(mv * mv, red, k);
  if (k == 0) memn[m] = sqrtf(s);
}

// ---- 6) precedence ----
__global__ void k_prec(const float* __restrict__ ww, const float* __restrict__ prec0,
                       float* __restrict__ prec) {
  __shared__ float red[1024];
  int w = blockIdx.x, tid = threadIdx.x;
  float lsum = 0.f;
  for (int j = 0; j < 4; j++) lsum += ww[(size_t)w * MSZ + tid + j * 1024];
  float gsum = blockSum<1024>(lsum, red, tid);
  for (int j = 0; j < 4; j++) {
    int m = tid + j * 1024;
    prec[(size_t)w * MSZ + m] = (1.f - gsum) * prec0[(size_t)w * MSZ + m] + ww[(size_t)w * MSZ + m];
  }
}

// ---- 7) link update fused with fw (row-parallel, exact) ----
__global__ void k_link_fw(const float* __restrict__ link0, const float* __restrict__ ww,
                          const float* __restrict__ prec0, const float* __restrict__ rw0,
                          float* __restrict__ link, float* __restrict__ fw) {
  __shared__ float red[256];
  int w = blockIdx.x >> 12, i = blockIdx.x & 4095, tid = threadIdx.x;
  float wwi = ww[(size_t)w * MSZ + i];
  const float* l0 = link0 + ((size_t)w * MSZ + i) * MSZ;
  float* lo = link + ((size_t)w * MSZ + i) * MSZ;
  const float* wwr = ww + (size_t)w * MSZ;
  const float* pr = prec0 + (size_t)w * MSZ;
  float f0 = 0.f, f1 = 0.f, f2 = 0.f, f3 = 0.f;
  for (int j = tid; j < MSZ; j += 256) {
    float lv = (1.f - wwi - wwr[j]) * l0[j] + wwi * pr[j];
    if (j == i) lv = 0.f;
    lo[j] = lv;
    f0 += rw0[j] * lv; f1 += rw0[MSZ + j] * lv;
    f2 += rw0[2 * MSZ + j] * lv; f3 += rw0[3 * MSZ + j] * lv;
  }
  float g;
  g = blockSum<256>(f0, red, tid); if (tid == 0) fw[((size_t)w * RH + 0) * MSZ + i] = g;
  g = blockSum<256>(f1, red, tid); if (tid == 0) fw[((size_t)w * RH + 1) * MSZ + i] = g;
  g = blockSum<256>(f2, red, tid); if (tid == 0) fw[((size_t)w * RH + 2) * MSZ + i] = g;
  g = blockSum<256>(f3, red, tid); if (tid == 0) fw[((size_t)w * RH + 3) * MSZ + i] = g;
}

// ---- zero helper ----
__global__ void k_zero(float* __restrict__ p, int n) {
  int i = blockIdx.x * 256 + threadIdx.x;
  if (i < n) p[i] = 0.f;
}

// ---- zero-pad a 4x4096 matrix to 16x4096 (rows 4..15 = 0) ----
__global__ void k_pad16(const float* __restrict__ src, float* __restrict__ dst) {
  int idx = blockIdx.x * 256 + threadIdx.x;   // 16*4096 elements
  int row = idx >> 12, m = idx & 4095;
  dst[idx] = (row < RH) ? src[(size_t)row * MSZ + m] : 0.f;
}

// ---- 8) bw = rw0 @ link  via V_WMMA_F32_16X16X4_F32 (exact f32) ----
// A operand comes from the zero-padded rw0p: unconditional, branch-free loads.
__global__ void k_bw(const float* __restrict__ link, const float* __restrict__ rw0p,
                     float* __restrict__ bw) {
  int wid = blockIdx.x * 8 + (threadIdx.x >> 5);     // 2048 waves
  int w = wid >> 10;
  int rem = wid & 1023;
  int ic = rem >> 8, jt = rem & 255;
  int j0 = jt * 16, i0 = ic * 1024;
  int lane = threadIdx.x & 31, half = lane >> 4, ln = lane & 15;
  const float* lk = link + (size_t)w * MSZ * MSZ;
  const float* arow = rw0p + (size_t)ln * MSZ;       // padded: rows >=4 are zero
#if HAVE_WMMA_F32X4
  v8f c = {};
  for (int i = i0; i < i0 + 1024; i += 4) {
    int kb = i + half * 2;                           // even -> 8B aligned
    v2f a = *(const v2f*)(arow + kb);
    v2f b;
    b.x = lk[(size_t)kb * MSZ + j0 + ln];
    b.y = lk[(size_t)(kb + 1) * MSZ + j0 + ln];
    c = __builtin_amdgcn_wmma_f32_16x16x4_f32(false, a, false, b, (short)0, c, false, false);
  }
  if (half == 0) {
    for (int v = 0; v < 4; v++)
      atomicAdd(&bw[((size_t)w * RH + v) * MSZ + j0 + ln], c[v]);
  }
#else
  if (half == 0) {
    float acc[4] = {0.f, 0.f, 0.f, 0.f};
    for (int i = i0; i < i0 + 1024; i++) {
      float lv = lk[(size_t)i * MSZ + j0 + ln];
      for (int r = 0; r < 4; r++) acc[r] += rw0p[(size_t)r * MSZ + i] * lv;
    }
    for (int r = 0; r < 4; r++)
      atomicAdd(&bw[((size_t)w * RH + r) * MSZ + j0 + ln], acc[r]);
  }
#endif
}

// ---- 9) dot_r (one wave per memory row) ----
__global__ void k_dotr(const float* __restrict__ memo, const float* __restrict__ proj,
                       float* __restrict__ dotr) {
  __shared__ float srk[512];
  int tid = threadIdx.x;
  srk[tid] = proj[794 + tid];
  srk[256 + tid] = proj[794 + 256 + tid];
  __syncthreads();
  int wave = tid >> 5, lane = tid & 31;
  int m = blockIdx.x * 8 + wave;
  const float* row = memo + (size_t)m * WORDL;
  float a0 = 0.f, a1 = 0.f, a2 = 0.f, a3 = 0.f;
  for (int k = lane; k < WORDL; k += 32) {
    float v = row[k];
    a0 += v * srk[k]; a1 += v * srk[128 + k]; a2 += v * srk[256 + k]; a3 += v * srk[384 + k];
  }
  a0 = waveSum(a0); a1 = waveSum(a1); a2 = waveSum(a2); a3 = waveSum(a3);
  if (lane == 0) { dotr[m] = a0; dotr[MSZ + m] = a1; dotr[2 * MSZ + m] = a2; dotr[3 * MSZ + m] = a3; }
}

// ---- 10) read-content softmax -> cw ----
__global__ void k_cw(const float* __restrict__ dotr, const float* __restrict__ memn,
                     const float* __restrict__ norms, const float* __restrict__ proj,
                     float* __restrict__ cw) {
  __shared__ float red[1024];
  int r = blockIdx.x, tid = threadIdx.x;
  float kn = norms[2 + r], rsr = proj[1306 + r];
  float s[4], lmax = -3.4e38f;
  for (int j = 0; j < 4; j++) {
    int m = tid + j * 1024;
    float sc = dotr[(size_t)r * MSZ + m] / (kn * memn[m]) * rsr;
    s[j] = sc; lmax = fmaxf(lmax, sc);
  }
  float gmax = blockMax<1024>(lmax, red, tid);
  float e[4], lsum = 0.f;
  for (int j = 0; j < 4; j++) { e[j] = expf(s[j] - gmax); lsum += e[j]; }
  float gsum = blockSum<1024>(lsum, red, tid);
  for (int j = 0; j < 4; j++) cw[(size_t)r * MSZ + tid + j * 1024] = e[j] / gsum;
}

// ---- 11) combine read weights ----
__global__ void k_rweights(const float* __restrict__ cw, const float* __restrict__ fw,
                           const float* __restrict__ bw, const float* __restrict__ proj,
                           float* __restrict__ rwt) {
  int idx = blockIdx.x * 256 + threadIdx.x;
  if (idx >= RH * MSZ) return;
  int r = idx >> 12, m = idx & 4095;
  const float* rm = proj + 516 + r * 5;   // [back0, back1, fwd0, fwd1, cont]
  float v = rm[4] * cw[idx]
          + rm[2] * fw[(size_t)(0 * RH + r) * MSZ + m]
          + rm[3] * fw[(size_t)(1 * RH + r) * MSZ + m]
          + rm[0] * bw[(size_t)(0 * RH + r) * MSZ + m]
          + rm[1] * bw[(size_t)(1 * RH + r) * MSZ + m];
  rwt[idx] = v;
}

// ---- 12) read_words = read_weights @ memory via WMMA f32 16x16x4 ----
__global__ void k_rwords(const float* __restrict__ rwtp, const float* __restrict__ memo,
                         float* __restrict__ outw) {
  int wid = blockIdx.x * 8 + (threadIdx.x >> 5);     // 32 waves
  int kt = wid & 7, mc = wid >> 3;
  int k0 = kt * 16, m0 = mc * 1024;
  int lane = threadIdx.x & 31, half = lane >> 4, ln = lane & 15;
  const float* arow = rwtp + (size_t)ln * MSZ;       // padded: rows >=4 are zero
#if HAVE_WMMA_F32X4
  v8f c = {};
  for (int m = m0; m < m0 + 1024; m += 4) {
    int kb = m + half * 2;
    v2f a = *(const v2f*)(arow + kb);
    v2f b;
    b.x = memo[(size_t)kb * WORDL + k0 + ln];
    b.y = memo[(size_t)(kb + 1) * WORDL + k0 + ln];
    c = __builtin_amdgcn_wmma_f32_16x16x4_f32(false, a, false, b, (short)0, c, false, false);
  }
  if (half == 0) {
    for (int v = 0; v < 4; v++)
      atomicAdd(&outw[v * WORDL + k0 + ln], c[v]);
  }
#else
  if (half == 0) {
    float acc[4] = {0.f, 0.f, 0.f, 0.f};
    for (int m = m0; m < m0 + 1024; m++) {
      float mv = memo[(size_t)m * WORDL + k0 + ln];
      for (int r = 0; r < 4; r++) acc[r] += rwtp[(size_t)r * MSZ + m] * mv;
    }
    for (int r = 0; r < 4; r++) atomicAdd(&outw[r * WORDL + k0 + ln], acc[r]);
  }
#endif
}

extern "C" void kernel_launch(void* const* d_in, const int* in_sizes, int n_in,
                              void* d_out, int out_size, void* d_ws, size_t ws_size,
                              hipStream_t stream) {
  (void)in_sizes; (void)n_in; (void)out_size; (void)ws_size;
  const float* x     = (const float*)d_in[0];
  const float* Wwv   = (const float*)d_in[1];  const float* bwv = (const float*)d_in[2];
  const float* Wev   = (const float*)d_in[3];  const float* bev = (const float*)d_in[4];
  const float* Wfg   = (const float*)d_in[5];  const float* bfg = (const float*)d_in[6];
  // d_in[7..10] = Wag/bag/Wwg/bwg : computed but unused by the reference outputs
  const float* Wrm   = (const float*)d_in[11]; const float* brm = (const float*)d_in[12];
  const float* Wwk   = (const float*)d_in[13]; const float* bwk = (const float*)d_in[14];
  const float* Wws   = (const float*)d_in[15]; const float* bws = (const float*)d_in[16];
  const float* Wrk   = (const float*)d_in[17]; const float* brk = (const float*)d_in[18];
  const float* Wrs   = (const float*)d_in[19]; const float* brs = (const float*)d_in[20];
  const float* mem0  = (const float*)d_in[21];
  const float* rw0   = (const float*)d_in[22];
  const float* ww0   = (const float*)d_in[23];
  const float* link0 = (const float*)d_in[24];
  const float* prec0 = (const float*)d_in[25];
  const float* usage0= (const float*)d_in[26];

  float* out = (float*)d_out;
  float* ws  = (float*)d_ws;
  float* proj  = ws + OFF_PROJ;
  float* norms = ws + OFF_NORMS;
  float* dotw  = ws + OFF_DOTW;
  float* an    = ws + OFF_AN;
  float* memn  = ws + OFF_MEMN;
  float* dotr  = ws + OFF_DOTR;
  float* cw    = ws + OFF_CW;
  float* fw    = ws + OFF_FW;
  float* bw    = ws + OFF_BW;
  float* rw0p  = ws + OFF_RW0P;
  float* rwtp  = ws + OFF_RWTP;

  // 0) projections (sigmoid where the reference applies it)
  k_linear<<<256, 256, 0, stream>>>(Wwv, bwv, x, proj + 0,    0);
  k_linear<<<256, 256, 0, stream>>>(Wev, bev, x, proj + 256,  1);
  k_linear<<<4,   256, 0, stream>>>(Wfg, bfg, x, proj + 512,  1);
  k_linear<<<20,  256, 0, stream>>>(Wrm, brm, x, proj + 516,  1);
  k_linear<<<256, 256, 0, stream>>>(Wwk, bwk, x, proj + 536,  1);
  k_linear<<<2,   256, 0, stream>>>(Wws, bws, x, proj + 792,  0);
  k_linear<<<512, 256, 0, stream>>>(Wrk, brk, x, proj + 794,  0);
  k_linear<<<4,   256, 0, stream>>>(Wrs, brs, x, proj + 1306, 0);
  k_norms<<<6, 128, 0, stream>>>(proj, norms);

  // zero accumulators + build padded rw0 (branch-free WMMA A-operand)
  k_zero<<<(NWW * RH * MSZ + 255) / 256, 256, 0, stream>>>(bw, NWW * RH * MSZ);
  k_zero<<<2, 256, 0, stream>>>(out + O_RWORDS, RH * WORDL);
  k_pad16<<<(16 * MSZ) / 256, 256, 0, stream>>>(rw0, rw0p);

  // 2) usage
  k_usage<<<16, 256, 0, stream>>>(usage0, ww0, rw0, proj, out + O_USAGE);

  // 3-4) write weights
  k_dotw<<<512, 256, 0, stream>>>(mem0, proj, dotw, an);
  k_ww<<<2, 1024, 0, stream>>>(dotw, an, norms, proj, out + O_WW);

  // 5) memory + row norms
  k_memory<<<MSZ, 128, 0, stream>>>(mem0, out + O_WW, proj, out + O_MEM, memn);

  // 6) precedence
  k_prec<<<2, 1024, 0, stream>>>(out + O_WW, prec0, out + O_PREC);

  // 7) link + fw (fused, the bandwidth-dominant pass)
  k_link_fw<<<NWW * MSZ, 256, 0, stream>>>(link0, out + O_WW, prec0, rw0,
                                           out + O_LINK, fw);

  // 8) bw via f32 WMMA (branch-free inner loop)
  k_bw<<<256, 256, 0, stream>>>(out + O_LINK, rw0p, bw);

  // 9-10) read content weights
  k_dotr<<<512, 256, 0, stream>>>(out + O_MEM, proj, dotr);
  k_cw<<<4, 1024, 0, stream>>>(dotr, memn, norms, proj, cw);

  // 11) combine read weights (also feeds the padded copy for WMMA)
  k_rweights<<<(RH * MSZ + 255) / 256, 256, 0, stream>>>(cw, fw, bw, proj, out + O_RW);
  k_pad16<<<(16 * MSZ) / 256, 256, 0, stream>>>(out + O_RW, rwtp);

  // 12) read_words via f32 WMMA
  k_rwords<<<4, 256, 0, stream>>>(rwtp, out + O_MEM, out + O_RWORDS);
}